// FSAS_12902081757735
// MI455X (gfx1250) — compile-verified
//
#include <hip/hip_runtime.h>

typedef __attribute__((ext_vector_type(2))) float v2f;
typedef __attribute__((ext_vector_type(8))) float v8f;
typedef __attribute__((ext_vector_type(4))) unsigned int v4u;
typedef __attribute__((ext_vector_type(4))) int v4i;
typedef __attribute__((ext_vector_type(8))) int v8i;

#define Cc     48
#define C2     96
#define C6     288
#define Hh     256
#define Ww     256
#define PSTR   112   // padded halo-position stride (7*16 WMMA N-tiles)
#define QSTR   64    // qkv position stride
#define ASTR   68    // attn/mod position stride (avoid 64-bank conflicts in LayerNorm)

// LDS float counts
#define LDS_X_N    (Cc  * PSTR)   //  5376
#define LDS_HID_N  (C6  * PSTR)   // 32256
#define LDS_QKV_N  (C6  * QSTR)   // 18432
#define LDS_ATT_N  (C2  * ASTR)   //  6528
#define LDS_TOTAL_N (LDS_X_N + LDS_HID_N + LDS_QKV_N + LDS_ATT_N)  // 62592 fl = 250368 B < 320KB

__global__ __launch_bounds__(256, 1)
void fsas_fused_kernel(const float* __restrict__ x,
                       const float* __restrict__ w_hidden,
                       const float* __restrict__ b_hidden,
                       const float* __restrict__ w_dw,
                       const float* __restrict__ b_dw,
                       const float* __restrict__ gamma,
                       const float* __restrict__ beta,
                       const float* __restrict__ w_out,
                       const float* __restrict__ b_out,
                       float* __restrict__ out)
{
    extern __shared__ float smem[];
    float* lds_x   = smem;                   // [48][112]  x halo (10x10 -> 100 cols, rest 0)
    float* lds_hid = lds_x   + LDS_X_N;      // [288][112] hidden (halo positions)
    float* lds_qkv = lds_hid + LDS_HID_N;    // [288][64]  q|k|v after depthwise
    float* lds_att = lds_qkv + LDS_QKV_N;    // [96][68]   circ-conv result, then normed*v

    const int tid  = threadIdx.x;
    const int lane = tid & 31;
    const int wave = tid >> 5;          // 0..7
    const int hi   = lane >> 4;         // lane group 0/1 (WMMA K/M split)
    const int lm   = lane & 15;         // 0..15

    const int wg    = blockIdx.x;       // b*1024 + pY*32 + pX
    const int b     = wg >> 10;
    const int pY    = (wg >> 5) & 31;
    const int pX    = wg & 31;
    const int y0    = pY * 8;
    const int x0    = pX * 8;

    if (tid == 0) {                     // warm weight lines into cache (global_prefetch_b8)
        __builtin_prefetch(w_hidden, 0, 1);
        __builtin_prefetch(w_out, 0, 1);
        __builtin_prefetch(w_dw, 0, 1);
    }

    // ---- Stage 1: x-halo (10x10, 48 ch) into LDS ----
    // Interior patches: Tensor Data Mover (iterate mode: one 10x10 2D tile per
    // channel, lds increment 112 elements, global increment H*W) — zero VALU
    // traffic, tracked on TENSORcnt. Border patches need SAME zero padding
    // *before* the tensor start, which TDM cannot express -> scalar path.
    const bool interior = (pY > 0) && (pY < 31) && (pX > 0) && (pX < 31);
#if __has_builtin(__builtin_amdgcn_tensor_load_to_lds)
    if (interior) {
        // zero the WMMA N-tile pad (cols 100..111) while TDM streams cols 0..99
        for (int i = tid; i < Cc * 12; i += 256) {
            const int c = i / 12, p = 100 + i % 12;
            lds_x[c * PSTR + p] = 0.0f;
        }
        if (wave == 0) {
            const unsigned long long gaddr =
                (unsigned long long)(const void*)(x + (((size_t)b * Cc) * Hh + (y0 - 1)) * Ww + (x0 - 1));
            const unsigned int lds_addr = (unsigned int)(unsigned long long)(const void*)lds_x;
            const unsigned int tdim0 = (unsigned int)(Ww - (x0 - 1));   // extent from tile start
            const unsigned int tdim1 = (unsigned int)(Hh - (y0 - 1));
            v4u g0;
            g0.x = 1u;                                            // count=1 (valid), user desc
            g0.y = lds_addr;                                      // LDS byte address
            g0.z = (unsigned int)(gaddr & 0xFFFFFFFFull);         // global_addr[31:0]
            g0.w = (unsigned int)((gaddr >> 32) & 0x01FFFFFFull)  // global_addr[56:32]
                   | (2u << 30);                                  // type=2 ("image")
            v8i g1;
            g1[0] = (int)((2u << 16) | (1u << 19));               // data_size=4B, iterate_enable
            g1[1] = (int)((tdim0 & 0xFFFFu) << 16);               // tensor_dim0[15:0]
            g1[2] = (int)(((tdim0 >> 16) & 0xFFFFu) | ((tdim1 & 0xFFFFu) << 16));
            g1[3] = (int)(((tdim1 >> 16) & 0xFFFFu) | (10u << 16)); // tile_dim0=10
            g1[4] = (int)(10u);                                   // tile_dim1=10, tile_dim2=0 (2D)
            g1[5] = (int)(256u);                                  // tensor_dim0_stride[31:0] = W
            g1[6] = 0;                                            // stride0[47:32]=0, stride1[15:0]=0
            g1[7] = 1;                                            // tensor_dim1_stride[47:16]: 65536>>16
            v4i g2;
            g2[0] = Cc;                                           // tensor_dim2 (unused in iterate)
            g2[1] = PSTR;                                         // lds_addr_increment (elements)
            g2[2] = Hh * Ww;                                      // global_addr_increment (elements)
            g2[3] = (int)((Cc - 1) << 16);                        // iterate_count = 47 -> 48 iters
            v4i g3 = {0, 0, 0, 0};
            v8i g4 = {0, 0, 0, 0, 0, 0, 0, 0};                    // extra words (clang-23 6-arg form)
            __builtin_amdgcn_tensor_load_to_lds(g0, g1, g2, g3, g4, 0);
            __builtin_amdgcn_s_wait_tensorcnt(0);
        }
    } else
#endif
    {
        for (int i = tid; i < Cc * PSTR; i += 256) {
            const int c = i / PSTR, p = i % PSTR;
            float v = 0.0f;
            if (p < 100) {
                const int ty = p / 10, tx = p % 10;
                const int gy = y0 - 1 + ty, gx = x0 - 1 + tx;
                if (gy >= 0 && gy < Hh && gx >= 0 && gx < Ww)
                    v = x[(((size_t)b * Cc + c) * Hh + gy) * Ww + gx];
            }
            lds_x[c * PSTR + p] = v;
        }
    }
    __syncthreads();

    // ---- Stage 2: hidden[288 x 100] = Wh[288 x 48] * x[48 x 100] via f32 WMMA ----
    // 18 M-tiles x 7 N-tiles, round-robin over 8 waves (wave-uniform -> EXEC all ones)
    for (int t = wave; t < 18 * 7; t += 8) {
        const int mt = t / 7, nt = t % 7;
        const int m = mt * 16 + lm;      // A row (lane = M)
        const int n = nt * 16 + lm;      // B col (lane = N)
        v8f acc = {};
        for (int k0 = 0; k0 < Cc; k0 += 4) {
            const int ka = k0 + 2 * hi;  // ISA 16x4 f32 A layout: lanes16-31 hold K+2
            v2f av, bv;
            av.x = w_hidden[m * Cc + ka];
            av.y = w_hidden[m * Cc + ka + 1];
            bv.x = lds_x[ka * PSTR + n];
            bv.y = lds_x[(ka + 1) * PSTR + n];
            acc = __builtin_amdgcn_wmma_f32_16x16x4_f32(
                false, av, false, bv, (short)0, acc, false, false);
        }
        // C/D layout: VGPR r holds row (r + 8*hi), col = lm; fold in bias
        #pragma unroll
        for (int r = 0; r < 8; ++r) {
            const int mr = mt * 16 + r + hi * 8;
            lds_hid[mr * PSTR + n] = acc[r] + b_hidden[mr];
        }
    }
    __syncthreads();

    // ---- Stage 3: depthwise 3x3 (SAME) over halo -> qkv[288 x 64] ----
    for (int i = tid; i < C6 * 64; i += 256) {
        const int ch = i >> 6, p = i & 63;
        const int py = p >> 3, px = p & 7;
        const float* wd   = &w_dw[ch * 9];
        const float* hrow = &lds_hid[ch * PSTR];
        float s = b_dw[ch];
        #pragma unroll
        for (int dy = 0; dy < 3; ++dy)
            #pragma unroll
            for (int dx = 0; dx < 3; ++dx)
                s += wd[dy * 3 + dx] * hrow[(py + dy) * 10 + (px + dx)];
        lds_qkv[ch * QSTR + p] = s;
    }
    __syncthreads();

    // ---- Stage 4: per-channel 8x8 circular conv == irfft2(rfft2(q)*rfft2(k)) ----
    for (int i = tid; i < C2 * 64; i += 256) {
        const int c = i >> 6, p = i & 63;
        const int yy = p >> 3, xx = p & 7;
        const float* q = &lds_qkv[c * QSTR];
        const float* k = &lds_qkv[(C2 + c) * QSTR];
        float s = 0.0f;
        #pragma unroll
        for (int u = 0; u < 8; ++u) {
            const int ry = (yy - u) & 7;
            #pragma unroll
            for (int v = 0; v < 8; ++v)
                s += q[u * 8 + v] * k[ry * 8 + ((xx - v) & 7)];
        }
        lds_att[c * ASTR + p] = s;
    }
    __syncthreads();

    // ---- Stage 5: LayerNorm over channels + gamma/beta + elementwise v (in place) ----
    if (tid < 64) {   // waves 0..1 fully active, 2..7 fully skip -> wave-uniform EXEC
        float sum = 0.0f, sq = 0.0f;
        for (int c = 0; c < C2; ++c) {
            const float t = lds_att[c * ASTR + tid];
            sum += t; sq += t * t;
        }
        const float mu  = sum * (1.0f / C2);
        const float var = sq * (1.0f / C2) - mu * mu;     // biased var (jnp.var)
        const float inv = rsqrtf(var + 1e-5f);
        for (int c = 0; c < C2; ++c) {
            const float nrm = (lds_att[c * ASTR + tid] - mu) * inv * gamma[c] + beta[c];
            lds_att[c * ASTR + tid] = nrm * lds_qkv[(2 * C2 + c) * QSTR + tid];
        }
    }
    __syncthreads();

    // ---- Stage 6: out[48 x 64] = Wout[48 x 96] * mod[96 x 64] via f32 WMMA ----
    for (int t = wave; t < 3 * 4; t += 8) {
        const int mt = t >> 2, nt = t & 3;
        const int m = mt * 16 + lm;
        const int n = nt * 16 + lm;       // position 0..63 within patch
        v8f acc = {};
        for (int k0 = 0; k0 < C2; k0 += 4) {
            const int ka = k0 + 2 * hi;
            v2f av, bv;
            av.x = w_out[m * C2 + ka];
            av.y = w_out[m * C2 + ka + 1];
            bv.x = lds_att[ka * ASTR + n];
            bv.y = lds_att[(ka + 1) * ASTR + n];
            acc = __builtin_amdgcn_wmma_f32_16x16x4_f32(
                false, av, false, bv, (short)0, acc, false, false);
        }
        #pragma unroll
        for (int r = 0; r < 8; ++r) {
            const int mr = mt * 16 + r + hi * 8;   // output channel 0..47
            const int gy = y0 + (n >> 3);
            const int gx = x0 + (n & 7);
            out[(((size_t)b * Cc + mr) * Hh + gy) * Ww + gx] = acc[r] + b_out[mr];
        }
    }
}

extern "C" void kernel_launch(void* const* d_in, const int* in_sizes, int n_in,
                              void* d_out, int out_size, void* d_ws, size_t ws_size,
                              hipStream_t stream)
{
    (void)in_sizes; (void)n_in; (void)d_ws; (void)ws_size; (void)out_size;
    const float* x        = (const float*)d_in[0];
    const float* w_hidden = (const float*)d_in[1];
    const float* b_hidden = (const float*)d_in[2];
    const float* w_dw     = (const float*)d_in[3];
    const float* b_dw     = (const float*)d_in[4];
    const float* gamma    = (const float*)d_in[5];
    const float* beta     = (const float*)d_in[6];
    const float* w_out    = (const float*)d_in[7];
    const float* b_out    = (const float*)d_in[8];
    float* out            = (float*)d_out;

    const size_t shmem = (size_t)LDS_TOTAL_N * sizeof(float);  // ~245 KB (CDNA5 WGP LDS = 320 KB)
    const dim3 grid(4 * 32 * 32);   // one workgroup per (batch, 8x8 patch)
    const dim3 block(256);
    fsas_fused_kernel<<<grid, block, shmem, stream>>>(
        x, w_hidden, b_hidden, w_dw, b_dw, gamma, beta, w_out, b_out, out);
}